// StructNConv2D_gx_with_ds_79396765434446
// MI455X (gfx1250) — compile-verified
//
#include <hip/hip_runtime.h>
#include <hip/hip_bf16.h>

#define EPSV 1e-20f
#define HH 256
#define WW 256
#define HS 254          // H - 2
#define HOUT 127        // ceil(254/2)
#define NTOT 32258      // B * 127 * 127
#define XSZ  1032256    // C * NTOT
#define OUT1 1032256    // B * O * 127 * 127

typedef __attribute__((ext_vector_type(2))) float v2f;
typedef __attribute__((ext_vector_type(8))) float v8f;

__device__ __forceinline__ float softplus_f(float x) {
    // jax.nn.softplus: max(x,0) + log1p(exp(-|x|))
    return fmaxf(x, 0.0f) + log1pf(__expf(-fabsf(x)) * 0.0f + expf(-fabsf(x)));
}

// ---------------------------------------------------------------------------
// Kernel 0: softplus all weights, compute their sums. One block.
// ws layout (floats): wsp[288], wch[1024], sums[2] = {sum_sw, sum_cw}
// ---------------------------------------------------------------------------
__global__ __launch_bounds__(256) void init_weights(
    const float* __restrict__ spatial_w, const float* __restrict__ channel_w,
    float* __restrict__ wsp, float* __restrict__ wch, float* __restrict__ sums) {
    __shared__ float r1[256], r2[256];
    const int tid = threadIdx.x;
    float psw = 0.0f, pcw = 0.0f;
    for (int i = tid; i < 288; i += 256) {
        float v = softplus_f(spatial_w[i]);
        wsp[i] = v; psw += v;
    }
    for (int i = tid; i < 1024; i += 256) {
        float v = softplus_f(channel_w[i]);
        wch[i] = v; pcw += v;
    }
    r1[tid] = psw; r2[tid] = pcw;
    __syncthreads();
    for (int s = 128; s > 0; s >>= 1) {
        if (tid < s) { r1[tid] += r1[tid + s]; r2[tid] += r2[tid + s]; }
        __syncthreads();
    }
    if (tid == 0) { sums[0] = r1[0]; sums[1] = r2[0]; }
}

// ---------------------------------------------------------------------------
// Kernel 1: fused elementwise (gx_f/cgx_f) + 3x3 strided spatial aggregation.
// One block = one (b,c) plane, 8x32 output tile. LDS halo 17x65 so each input
// element is computed/loaded once instead of 9 times.
// Writes X1 = cgx_sp*gx_sp and X2 = cgx_sp in [c][n] layout for the GEMM.
// ---------------------------------------------------------------------------
#define TTH 8
#define TTW 32
#define RH 17           // 2*TTH + 1
#define RW 65           // 2*TTW + 1

__global__ __launch_bounds__(256) void spatial_kernel(
    const float* __restrict__ dP0, const float* __restrict__ cdP0,
    const float* __restrict__ sP0, const float* __restrict__ gxP0,
    const float* __restrict__ cgxP0, const float* __restrict__ spr,
    const float* __restrict__ w_prop, const float* __restrict__ wsp,
    const float* __restrict__ sums,
    float* __restrict__ X1, float* __restrict__ X2) {
    __shared__ float sg[RH][RW];   // gx_f halo tile
    __shared__ float sc[RH][RW];   // cgx_f halo tile

    const int tileW = blockIdx.x & 3;
    const int tileH = blockIdx.x >> 2;
    const int c = blockIdx.y, b = blockIdx.z;
    const int hs0 = tileH * TTH, ws0 = tileW * TTW;
    const int tid = threadIdx.x;

    const float wp = softplus_f(w_prop[0]);
    const size_t plane = (size_t)(b * 32 + c) * (HH * WW);
    const float* __restrict__ dP  = dP0  + plane;
    const float* __restrict__ cdP = cdP0 + plane;
    const float* __restrict__ sP  = sP0  + plane;
    const float* __restrict__ gxP = gxP0 + plane;
    const float* __restrict__ cgP = cgxP0 + plane;

    // ---- stage 1: build gx_f / cgx_f halo region in LDS ----
    for (int idx = tid; idx < RH * RW; idx += 256) {
        const int rh = idx / RW, rw = idx - rh * RW;
        int hh = 2 * hs0 + rh; if (hh > HH - 1) hh = HH - 1;
        int wx = 2 * ws0 + rw; if (wx > WW - 1) wx = WW - 1;
        const int wl = (wx == 0) ? WW - 1 : wx - 1;       // roll wrap
        const int wr = (wx == WW - 1) ? 0 : wx + 1;
        const int row = hh * WW;
        const float cdl = (wx == 0) ? 0.0f : cdP[row + wx - 1];
        const float cdr = (wx == WW - 1) ? 0.0f : cdP[row + wx + 1];
        const float dl = dP[row + wl], dr = dP[row + wr];
        const float sl = sP[row + wl], sr = sP[row + wr];
        const float scv = sP[row + wx];
        const float gv = gxP[row + wx], cgv = cgP[row + wx];

        const float cgx_ds = scv * sl * sr * cdl * cdr;
        const float height = (cdl * dl + cdr * dr) / (cdl + cdr + EPSV);
        const float gx_ds  = (dr - dl) * 0.5f / (height + EPSV);
        const float num    = wp * cgv * gv + cgx_ds * gx_ds;
        const float den    = wp * cgv + cgx_ds;
        sg[rh][rw] = num / (den + EPSV);
        sc[rh][rw] = den / (wp + 1.0f);
    }
    __syncthreads();

    // ---- stage 2: 3x3 weighted aggregation, stride-2 sampled ----
    const int th = tid >> 5, tw = tid & 31;
    const int hs = hs0 + th, ws = ws0 + tw;
    if (hs < HOUT && ws < HOUT) {
        const float sumsw = sums[0];
        const size_t sprBase = (size_t)(b * 32 + c) * 9 * (HS * HS);
        float nom = 0.0f, den = 0.0f;
#pragma unroll
        for (int m = 0; m < 9; ++m) {
            const int i = m / 3, j = m - 3 * i;
            const float g  = sg[2 * th + i][2 * tw + j];
            const float cg = sc[2 * th + i][2 * tw + j];
            const float sv = spr[sprBase + (size_t)m * (HS * HS)
                                 + (2 * hs + i) * HS + (2 * ws + j)];
            const float cp = cg * sv;
            const float wm = wsp[c * 9 + m];
            nom = fmaf(wm * cp, g, nom);
            den = fmaf(wm, cp, den);
        }
        const float gx_sp  = nom / (den + EPSV);
        const float cgx_sp = den / sumsw;
        const int n = b * (HOUT * HOUT) + hs * HOUT + ws;
        X1[(size_t)c * NTOT + n] = cgx_sp * gx_sp;
        X2[(size_t)c * NTOT + n] = cgx_sp;
    }
}

// ---------------------------------------------------------------------------
// Kernel 2: channel mix via V_WMMA_F32_16X16X4_F32.
//   Y[o,n] = sum_c Wch[o,c] * X[c,n]   (O=C=32, N=32258), for X1 and X2.
// One wave handles a 16-wide N strip: 2 O-tiles x 2 accumulators,
// K-loop of 8 (K=4 each) -> 32 WMMAs per wave, full f32 accumulation.
// Fragment layouts per ISA 7.12.2:
//   A 16x4 (MxK): lane M = lane&15, VGPR v holds K = v + 2*(lane>>4)
//   B 4x16 (KxN): lane N = lane&15, VGPR v holds K = v + 2*(lane>>4)
//   C/D 16x16:    VGPR r -> M = r + 8*(lane>>4), N = lane&15
// ---------------------------------------------------------------------------
__global__ __launch_bounds__(256) void channel_mix_wmma(
    const float* __restrict__ X1, const float* __restrict__ X2,
    const float* __restrict__ Wch, const float* __restrict__ sums,
    const float* __restrict__ bias, float* __restrict__ out) {
    const int lane = threadIdx.x & 31;
    const int wave = threadIdx.x >> 5;
    const int n0 = (blockIdx.x * 8 + wave) * 16;
    if (n0 >= NTOT) return;

    const int m  = lane & 15;           // A row / D column index within tile
    const int kh = (lane >> 4) << 1;    // lane-half K offset: 0 or 2
    const int n  = n0 + m;
    const int nc = (n < NTOT) ? n : (NTOT - 1);  // clamp so EXEC stays full

    v8f accN0 = {}, accD0 = {}, accN1 = {}, accD1 = {};
#pragma unroll
    for (int k0 = 0; k0 < 32; k0 += 4) {
        const int ka = k0 + kh;
        v2f a0, a1, b1, b2;
        a0.x = Wch[m * 32 + ka];        a0.y = Wch[m * 32 + ka + 1];
        a1.x = Wch[(m + 16) * 32 + ka]; a1.y = Wch[(m + 16) * 32 + ka + 1];
        b1.x = X1[(size_t)ka * NTOT + nc];       b1.y = X1[(size_t)(ka + 1) * NTOT + nc];
        b2.x = X2[(size_t)ka * NTOT + nc];       b2.y = X2[(size_t)(ka + 1) * NTOT + nc];
        accN0 = __builtin_amdgcn_wmma_f32_16x16x4_f32(false, a0, false, b1,
                                                      (short)0, accN0, false, false);
        accD0 = __builtin_amdgcn_wmma_f32_16x16x4_f32(false, a0, false, b2,
                                                      (short)0, accD0, false, false);
        accN1 = __builtin_amdgcn_wmma_f32_16x16x4_f32(false, a1, false, b1,
                                                      (short)0, accN1, false, false);
        accD1 = __builtin_amdgcn_wmma_f32_16x16x4_f32(false, a1, false, b2,
                                                      (short)0, accD1, false, false);
    }

    if (n >= NTOT) return;              // mask tail lanes (after all WMMAs)

    const float inv4cw = 1.0f / (4.0f * sums[1]);
    const int b = n / (HOUT * HOUT);
    const int rem = n - b * (HOUT * HOUT);
    const int hs = rem / HOUT;
    const int wsp_ = rem - hs * HOUT;
    const int mbase = (lane >> 4) * 8;

#pragma unroll
    for (int r = 0; r < 8; ++r) {
        // O-tile 0 (o = 0..15)
        int o = mbase + r;
        float nom = accN0[r], den = accD0[r];
        size_t idx = ((size_t)(b * 32 + o) * HOUT + hs) * HOUT + wsp_;
        out[idx]        = (nom / (den + EPSV) + bias[o]) * 2.0f;  // gx_o * STRIDE
        out[OUT1 + idx] = den * inv4cw;                           // cgx_o / 4
        // O-tile 1 (o = 16..31)
        o = mbase + r + 16;
        nom = accN1[r]; den = accD1[r];
        idx = ((size_t)(b * 32 + o) * HOUT + hs) * HOUT + wsp_;
        out[idx]        = (nom / (den + EPSV) + bias[o]) * 2.0f;
        out[OUT1 + idx] = den * inv4cw;
    }
}

// ---------------------------------------------------------------------------
extern "C" void kernel_launch(void* const* d_in, const int* in_sizes, int n_in,
                              void* d_out, int out_size, void* d_ws, size_t ws_size,
                              hipStream_t stream) {
    const float* d      = (const float*)d_in[0];
    const float* cd     = (const float*)d_in[1];
    const float* s      = (const float*)d_in[2];
    // d_in[3] = cs    : unused by gx variant
    const float* gx     = (const float*)d_in[4];
    const float* cgx    = (const float*)d_in[5];
    // d_in[6] = gy, d_in[7] = cgy : unused
    const float* spr    = (const float*)d_in[8];
    const float* w_prop = (const float*)d_in[9];
    const float* spw    = (const float*)d_in[10];
    const float* chw    = (const float*)d_in[11];
    const float* bias   = (const float*)d_in[12];
    float* out = (float*)d_out;

    float* ws   = (float*)d_ws;
    float* X1   = ws;                  // [C][NTOT]
    float* X2   = X1 + XSZ;            // [C][NTOT]
    float* wsp  = X2 + XSZ;            // [288]
    float* wch  = wsp + 288;           // [1024]
    float* sums = wch + 1024;          // [2]

    init_weights<<<1, 256, 0, stream>>>(spw, chw, wsp, wch, sums);

    dim3 gridA(64, 32, 2);             // 16x4 tiles, C=32, B=2
    spatial_kernel<<<gridA, 256, 0, stream>>>(d, cd, s, gx, cgx, spr,
                                              w_prop, wsp, sums, X1, X2);

    const int nWaves  = (NTOT + 15) / 16;      // 2017
    const int nBlocks = (nWaves + 7) / 8;      // 253
    channel_mix_wmma<<<nBlocks, 256, 0, stream>>>(X1, X2, wch, sums, bias, out);
}